// Aggregator_3582002725211
// MI455X (gfx1250) — compile-verified
//
#include <hip/hip_runtime.h>

typedef __attribute__((ext_vector_type(2))) float v2f;
typedef __attribute__((ext_vector_type(8))) float v8f;

#define D 128
#define NEG_SLOPE 0.01f
#define LDS_STRIDE 132   // 128 + 4 pad: rows land 4 banks apart -> conflict-free b64 reads

// ---------------- zero the side accumulator ----------------
__global__ void ngcf_zero_kernel(float4* __restrict__ p, int n4) {
    int i = blockIdx.x * blockDim.x + threadIdx.x;
    if (i < n4) p[i] = make_float4(0.f, 0.f, 0.f, 0.f);
}

// ---------------- edge scatter: side[dst] += w * ego[src] ----------------
// One wave per edge: edge indices are wave-uniform (scalar loads), each lane
// handles a float4 chunk of the 128-wide row. Coalesced 512B row read,
// 4x global_atomic_add_f32 per lane (no return).
__global__ void ngcf_scatter_kernel(const float* __restrict__ ego,
                                    const float* __restrict__ ew,
                                    const int* __restrict__ esrc,
                                    const int* __restrict__ edst,
                                    float* __restrict__ side, int E) {
    int tid = blockIdx.x * blockDim.x + threadIdx.x;
    int e = tid >> 5;
    int c = tid & 31;
    if (e >= E) return;
    int s = esrc[e];
    int d = edst[e];
    float w = ew[e];
    float4 v = ((const float4*)(ego + (size_t)s * D))[c];
    float* dst = side + (size_t)d * D + c * 4;
    atomicAdd(dst + 0, w * v.x);
    atomicAdd(dst + 1, w * v.y);
    atomicAdd(dst + 2, w * v.z);
    atomicAdd(dst + 3, w * v.w);
}

// ---------------- fused dual-GEMM + bias + LeakyReLU + sum ----------------
// Block = 8 waves; wave w owns output columns [16w, 16w+16).
// B-fragments (W1/W2 slices) live in registers for the whole grid-stride loop.
// A-fragments come from an LDS tile of x1 = ego+side, x2 = ego*side.
__global__ void __launch_bounds__(256)
ngcf_gemm_kernel(const float* __restrict__ ego, const float* __restrict__ side,
                 const float* __restrict__ W1, const float* __restrict__ b1,
                 const float* __restrict__ W2, const float* __restrict__ b2,
                 float* __restrict__ out, int n_nodes, int ntiles) {
    __shared__ __align__(16) float x1[16 * LDS_STRIDE];
    __shared__ __align__(16) float x2[16 * LDS_STRIDE];

    const int lane  = threadIdx.x & 31;
    const int wave  = threadIdx.x >> 5;          // 0..7 -> output n-tile
    const int ncol  = wave * 16 + (lane & 15);   // this lane's output column
    const int khalf = (lane >> 4) * 2;           // K sub-offset per A/B layout

    // Preload this wave's 128x16 slices of W1/W2 as WMMA B-fragments.
    // B 4x16 layout: VGPR0 = rows {K, K+2} (lane halves), VGPR1 = rows {K+1, K+3}.
    v2f bf1[32], bf2[32];
#pragma unroll
    for (int kk = 0; kk < 32; ++kk) {
        int kr = kk * 4 + khalf;
        bf1[kk].x = W1[(size_t)kr * D + ncol];
        bf1[kk].y = W1[(size_t)(kr + 1) * D + ncol];
        bf2[kk].x = W2[(size_t)kr * D + ncol];
        bf2[kk].y = W2[(size_t)(kr + 1) * D + ncol];
    }
    const float bias1 = b1[ncol];
    const float bias2 = b2[ncol];

    for (int t = blockIdx.x; t < ntiles; t += gridDim.x) {
        // Cooperative build of the 16x128 tile of x1/x2 in LDS.
        {
            int r  = threadIdx.x >> 4;           // 0..15 (tile row)
            int c0 = (threadIdx.x & 15) * 8;     // 8 contiguous cols per thread
            int node = t * 16 + r;
            if (node >= n_nodes) node = n_nodes - 1;   // clamp (tail safety)
            const float* er = ego  + (size_t)node * D + c0;
            const float* sr = side + (size_t)node * D + c0;
#pragma unroll
            for (int j = 0; j < 8; ++j) {
                float e = er[j], s = sr[j];
                x1[r * LDS_STRIDE + c0 + j] = e + s;
                x2[r * LDS_STRIDE + c0 + j] = e * s;
            }
        }
        __syncthreads();

        v8f acc1 = {};
        v8f acc2 = {};
        const int arow = lane & 15;
#pragma unroll
        for (int kk = 0; kk < 32; ++kk) {
            // A 16x4 layout: lanes 0-15 hold {K,K+1}, lanes 16-31 hold {K+2,K+3}.
            int kc = kk * 4 + khalf;
            v2f a1 = *(const v2f*)&x1[arow * LDS_STRIDE + kc];
            v2f a2 = *(const v2f*)&x2[arow * LDS_STRIDE + kc];
            acc1 = __builtin_amdgcn_wmma_f32_16x16x4_f32(
                false, a1, false, bf1[kk], (short)0, acc1, false, false);
            acc2 = __builtin_amdgcn_wmma_f32_16x16x4_f32(
                false, a2, false, bf2[kk], (short)0, acc2, false, false);
        }

        // C/D layout: VGPR v, lanes 0-15 -> row v, lanes 16-31 -> row v+8.
#pragma unroll
        for (int v = 0; v < 8; ++v) {
            int row  = v + ((lane >> 4) << 3);
            int node = t * 16 + row;
            if (node < n_nodes) {
                float s1 = acc1[v] + bias1;
                float s2 = acc2[v] + bias2;
                s1 = s1 > 0.f ? s1 : NEG_SLOPE * s1;
                s2 = s2 > 0.f ? s2 : NEG_SLOPE * s2;
                out[(size_t)node * D + ncol] = s1 + s2;
            }
        }
        __syncthreads();
    }
}

extern "C" void kernel_launch(void* const* d_in, const int* in_sizes, int n_in,
                              void* d_out, int out_size, void* d_ws, size_t ws_size,
                              hipStream_t stream) {
    const float* ego = (const float*)d_in[0];
    const float* ew  = (const float*)d_in[1];
    const float* W1  = (const float*)d_in[2];
    const float* b1  = (const float*)d_in[3];
    const float* W2  = (const float*)d_in[4];
    const float* b2  = (const float*)d_in[5];
    const int* esrc  = (const int*)d_in[6];
    const int* edst  = (const int*)d_in[7];
    float* out  = (float*)d_out;
    float* side = (float*)d_ws;                 // N*128 floats of scratch

    const int N = in_sizes[0] / D;              // 100000
    const int E = in_sizes[1];                  // 1600000

    // 1) side = 0
    int n4 = (N * D) / 4;
    ngcf_zero_kernel<<<(n4 + 255) / 256, 256, 0, stream>>>((float4*)side, n4);

    // 2) side[dst] += w * ego[src]   (one wave per edge)
    long long sthreads = (long long)E * 32;
    int sblocks = (int)((sthreads + 255) / 256);
    ngcf_scatter_kernel<<<sblocks, 256, 0, stream>>>(ego, ew, esrc, edst, side, E);

    // 3) fused dual GEMM + LeakyReLU + sum (persistent blocks, 16-node tiles)
    int ntiles = (N + 15) / 16;                 // 6250
    int gblocks = ntiles < 2048 ? ntiles : 2048;
    ngcf_gemm_kernel<<<gblocks, 256, 0, stream>>>(ego, side, W1, b1, W2, b2,
                                                  out, N, ntiles);
}